// CSA_14800457302208
// MI455X (gfx1250) — compile-verified
//
#include <hip/hip_runtime.h>

typedef float v2f __attribute__((ext_vector_type(2)));
typedef float v8f __attribute__((ext_vector_type(8)));

constexpr int B_ = 8, C1_ = 128, C2_ = 256, N_ = 1024;
constexpr int HEADS_ = 8, KC_ = 3, D_ = 32;
constexpr float SCALE_ = 0.17677669529663687f; // 32^-0.5

#define DEV __device__ __forceinline__

DEV v8f wmma4(v2f a, v2f b, v8f c) {
  // V_WMMA_F32_16X16X4_F32 : D = A(16x4) * B(4x16) + C(16x16), fp32 exact
  return __builtin_amdgcn_wmma_f32_16x16x4_f32(false, a, false, b, (short)0, c,
                                               false, false);
}

// B-style fragment: element(vgpr j, lane l) = P[(2*(l>>4)+j)*ld + (l&15)]
// (also the A fragment of a matrix stored transposed: A[m][k] = P[k][m])
DEV v2f ldB(const float* P, int ld) {
  int l = threadIdx.x & 31;
  const float* p = P + ((l >> 4) << 1) * ld + (l & 15);
  v2f r; r.x = p[0]; r.y = p[ld];
  return r;
}

// A fragment for row-major A[m][k]: element(vgpr j, lane l) = A[(l&15)*ld + 2*(l>>4)+j]
DEV v2f ldA(const float* P, int ld) {
  int l = threadIdx.x & 31;
  const float* p = P + (l & 15) * ld + ((l >> 4) << 1);
  v2f r; r.x = p[0]; r.y = p[1];
  return r;
}

// gfx1250 async global->LDS copy (ASYNCcnt-tracked, no VGPR bounce).
// Low 32 bits of a generic pointer to __shared__ are the LDS byte offset.
DEV unsigned lds_off(const void* p) { return (unsigned)(uintptr_t)p; }

DEV void async_b64(unsigned lds_byte_off, const void* gaddr) {
  asm volatile("global_load_async_to_lds_b64 %0, %1, off"
               :: "v"(lds_byte_off), "v"(gaddr)
               : "memory");
}

DEV void wait_async0() {
  asm volatile("s_wait_asynccnt 0x0" ::: "memory");
}

// ---------------------------------------------------------------------------
// Generic per-batch GEMM: Out[b] = W (OxC) @ X[b] (CxN) (+ Add[b])
// One wave computes a 32(O) x 64(N) tile: two A frags share each B frag.
// ---------------------------------------------------------------------------
__global__ void __launch_bounds__(256)
gemm_wmma(const float* __restrict__ W, const float* __restrict__ X,
          float* __restrict__ Out, const float* __restrict__ Add,
          int O, int C, int N, int tilesO, int tilesN) {
  int wave = (blockIdx.x * blockDim.x + threadIdx.x) >> 5;
  int tpb = tilesO * tilesN;
  int b = wave / tpb;
  int t = wave - b * tpb;
  int om0 = (t % tilesO) * 32;
  int n0 = (t / tilesO) * 64;

  const float* Xb = X + (size_t)b * C * N;
  float* Ob = Out + (size_t)b * O * N;

  v8f acc[2][4] = {};
  for (int k = 0; k < C; k += 4) {
    v2f a0 = ldA(W + (size_t)om0 * C + k, C);
    v2f a1 = ldA(W + (size_t)(om0 + 16) * C + k, C);
#pragma unroll
    for (int j = 0; j < 4; ++j) {
      v2f bb = ldB(Xb + (size_t)k * N + n0 + j * 16, N);
      acc[0][j] = wmma4(a0, bb, acc[0][j]);
      acc[1][j] = wmma4(a1, bb, acc[1][j]);
    }
  }

  int l = threadIdx.x & 31;
  int half = l >> 4, lane16 = l & 15;
  const float* Ab = Add ? Add + (size_t)b * O * N : nullptr;
#pragma unroll
  for (int at = 0; at < 2; ++at)
#pragma unroll
    for (int j = 0; j < 4; ++j)
#pragma unroll
      for (int r = 0; r < 8; ++r) {
        int m = om0 + at * 16 + r + half * 8;
        int n = n0 + j * 16 + lane16;
        float v = acc[at][j][r];
        if (Ab) v += Ab[(size_t)m * N + n];
        Ob[(size_t)m * N + n] = v;
      }
}

// ---------------------------------------------------------------------------
// Cluster probabilities: logits[k] = W_cl[k,:] . x1[b,:,n] + b_cl[k]; softmax
// ---------------------------------------------------------------------------
__global__ void __launch_bounds__(256)
prob_kernel(const float* __restrict__ x1, const float* __restrict__ Wc,
            const float* __restrict__ bc, float* __restrict__ prob) {
  int idx = blockIdx.x * blockDim.x + threadIdx.x; // b*N + n
  int b = idx >> 10;
  int n = idx & (N_ - 1);
  const float* xb = x1 + (size_t)b * C2_ * N_ + n;
  float l0 = bc[0], l1 = bc[1], l2 = bc[2];
  for (int c = 0; c < C2_; ++c) {
    float xv = xb[(size_t)c * N_];
    l0 = fmaf(Wc[c], xv, l0);
    l1 = fmaf(Wc[C2_ + c], xv, l1);
    l2 = fmaf(Wc[2 * C2_ + c], xv, l2);
  }
  float mx = fmaxf(l0, fmaxf(l1, l2));
  float e0 = __expf(l0 - mx), e1 = __expf(l1 - mx), e2 = __expf(l2 - mx);
  float inv = 1.f / (e0 + e1 + e2);
  float* pb = prob + (size_t)b * KC_ * N_ + n;
  pb[0] = e0 * inv;
  pb[N_] = e1 * inv;
  pb[2 * N_] = e2 * inv;
}

// ---------------------------------------------------------------------------
// Cluster-masked attention, flash-style over S^T, S computed once per tile and
// reused for KC=3 clusters. Block = 8 waves; wave owns 16 query columns.
// S^T layout puts the softmax axis (m) on the register axis: reductions are
// 7 VALU + 1 shfl_xor(16); flash state is 3 scalars; output stores coalesced.
// K/V tiles double-buffered in LDS via async global->LDS copies (ASYNCcnt).
// ---------------------------------------------------------------------------
__global__ void __launch_bounds__(256)
attn_kernel(const float* __restrict__ qkv, const float* __restrict__ prob,
            float* __restrict__ accb) {
  __shared__ float lds_k[2][32 * 16]; // K^T tile [d][m]
  __shared__ float lds_v[2][32 * 17]; // V^T tile [d][m], padded rows

  int qb = blockIdx.x & 7;
  int bh = blockIdx.x >> 3;
  int b = bh >> 3;
  int h = bh & 7;
  const float* Qb = qkv + ((size_t)b * 3 * C2_ + h * D_) * N_;
  const float* Kb = qkv + ((size_t)b * 3 * C2_ + C2_ + h * D_) * N_;
  const float* Vb = qkv + ((size_t)b * 3 * C2_ + 2 * C2_ + h * D_) * N_;
  const float* pr = prob + (size_t)b * KC_ * N_;
  float* Ob = accb + ((size_t)b * C2_ + h * D_) * N_;

  int wid = threadIdx.x >> 5;
  int l = threadIdx.x & 31;
  int half = l >> 4, lane16 = l & 15;
  int n0 = qb * 128 + wid * 16;

  // Preload Q fragments (B operand of S^T = K @ Q^T) — reused 64x.
  v2f bq[8];
#pragma unroll
  for (int kk = 0; kk < 8; ++kk)
    bq[kk] = ldB(Qb + (size_t)(kk * 4) * N_ + n0, N_);

  // Per-column (query) cluster probs, SCALE folded in.
  float anl[KC_];
#pragma unroll
  for (int c = 0; c < KC_; ++c)
    anl[c] = pr[(size_t)c * N_ + n0 + lane16] * SCALE_;

  float rmax[KC_], rsum[KC_];
  v8f oacc[KC_][2] = {};
#pragma unroll
  for (int c = 0; c < KC_; ++c) { rmax[c] = -1e30f; rsum[c] = 0.f; }

  // Async stage of one 32(d) x 16(m) K and V tile into LDS buffer `buf`.
  auto stage = [&](int m0, int buf) {
    int i = (int)threadIdx.x;
    int d = i >> 3;         // 0..31
    int mo = (i & 7) * 2;   // element pair within the 16-wide row
    async_b64(lds_off(&lds_k[buf][d * 16 + mo]), Kb + (size_t)d * N_ + m0 + mo);
    async_b64(lds_off(&lds_v[buf][d * 17 + mo]), Vb + (size_t)d * N_ + m0 + mo);
  };

  stage(0, 0);
  wait_async0();
  __syncthreads();

  for (int it = 0; it < N_ / 16; ++it) {
    int buf = it & 1;
    int m0 = it * 16;
    if (it + 1 < N_ / 16) stage(m0 + 16, buf ^ 1); // prefetch next tile

    const float* KK = lds_k[buf];
    const float* VV = lds_v[buf];

    // S^T tile (16m x 16n) = K(16x32) @ Q^T(32x16), computed ONCE.
    v8f st = {};
#pragma unroll
    for (int kk = 0; kk < 8; ++kk) {
      v2f ak = ldB(&KK[kk * 4 * 16], 16);
      st = wmma4(ak, bq[kk], st);
    }
    // st[r] at lane l = S[n = n0+lane16][m = m0 + r + 8*half]

#pragma unroll
    for (int c = 0; c < KC_; ++c) {
      float pm = pr[(size_t)c * N_ + m0 + lane16]; // key prob, m = lane16
      float bmv[8];
#pragma unroll
      for (int r = 0; r < 8; ++r)
        bmv[r] = __shfl(pm, r + (half << 3), 32);  // prob for m = r+8*half

      v8f t;
#pragma unroll
      for (int r = 0; r < 8; ++r) t[r] = st[r] * (anl[c] * bmv[r]);

      // column (= query-row) max over the 16 m values of this tile
      float lm = t[0];
#pragma unroll
      for (int r = 1; r < 8; ++r) lm = fmaxf(lm, t[r]);
      lm = fmaxf(lm, __shfl_xor(lm, 16, 32));

      float nm = fmaxf(rmax[c], lm);
      float corr = __expf(rmax[c] - nm);
      rmax[c] = nm;

      v8f p;
      float ls = 0.f;
#pragma unroll
      for (int r = 0; r < 8; ++r) { p[r] = __expf(t[r] - nm); ls += p[r]; }
      ls += __shfl_xor(ls, 16, 32);
      rsum[c] = rsum[c] * corr + ls;

#pragma unroll
      for (int dt = 0; dt < 2; ++dt)
#pragma unroll
        for (int r = 0; r < 8; ++r) oacc[c][dt][r] *= corr;

      // O^T += V^T @ P^T.  B frags of P^T built from registers via shuffles:
      // need P^T[mreq][n=l&15]; source lane = (l&15)+16*(mreq>>3), reg mreq&7.
#pragma unroll
      for (int kk = 0; kk < 4; ++kk) {
        v2f bp;
        {
          const int ma = 4 * kk + 0, mb = 4 * kk + 2;
          float va = __shfl(p[ma & 7], lane16 + ((ma >> 3) << 4), 32);
          float vb = __shfl(p[mb & 7], lane16 + ((mb >> 3) << 4), 32);
          bp.x = half ? vb : va;
        }
        {
          const int ma = 4 * kk + 1, mb = 4 * kk + 3;
          float va = __shfl(p[ma & 7], lane16 + ((ma >> 3) << 4), 32);
          float vb = __shfl(p[mb & 7], lane16 + ((mb >> 3) << 4), 32);
          bp.y = half ? vb : va;
        }
#pragma unroll
        for (int dt = 0; dt < 2; ++dt) {
          v2f av = ldA(&VV[(dt * 16) * 17 + kk * 4], 17);
          oacc[c][dt] = wmma4(av, bp, oacc[c][dt]); // O = O*corr + P@V
        }
      }
    }

    wait_async0();   // our prefetch of tile it+1 has landed in LDS
    __syncthreads(); // everyone done with buf, everyone's prefetch visible
  }

  // out[d][n] = (1/KC) * sum_c O_c/l_c ; O^T layout -> coalesced stores
  float inv[KC_];
#pragma unroll
  for (int c = 0; c < KC_; ++c) inv[c] = 1.0f / rsum[c];
#pragma unroll
  for (int dt = 0; dt < 2; ++dt)
#pragma unroll
    for (int r = 0; r < 8; ++r) {
      float v = 0.f;
#pragma unroll
      for (int c = 0; c < KC_; ++c) v += oacc[c][dt][r] * inv[c];
      v *= (1.0f / KC_);
      int d = dt * 16 + r + half * 8;
      Ob[(size_t)d * N_ + n0 + lane16] = v;
    }
}

// ---------------------------------------------------------------------------
extern "C" void kernel_launch(void* const* d_in, const int* in_sizes, int n_in,
                              void* d_out, int out_size, void* d_ws,
                              size_t ws_size, hipStream_t stream) {
  (void)in_sizes; (void)n_in; (void)out_size; (void)ws_size;
  const float* x      = (const float*)d_in[0];
  const float* W_in   = (const float*)d_in[1];
  const float* W_cl   = (const float*)d_in[2];
  const float* b_cl   = (const float*)d_in[3];
  const float* W_qkv  = (const float*)d_in[4];
  const float* W_proj = (const float*)d_in[5];
  float* out = (float*)d_out;

  char* ws = (char*)d_ws;
  float* x1   = (float*)(ws);                                  // 8 MB
  float* qkvb = (float*)(ws + (size_t)8 * 1024 * 1024);        // 24 MB
  float* accb = (float*)(ws + (size_t)32 * 1024 * 1024);       // 8 MB
  float* prob = (float*)(ws + (size_t)40 * 1024 * 1024);       // 96 KB

  // 1) proj_in: x1 = W_in @ x
  {
    int O = C2_, C = C1_, tO = O / 32, tN = N_ / 64;
    gemm_wmma<<<B_ * tO * tN / 8, 256, 0, stream>>>(W_in, x, x1, nullptr,
                                                    O, C, N_, tO, tN);
  }
  // 2) cluster probabilities
  prob_kernel<<<(B_ * N_) / 256, 256, 0, stream>>>(x1, W_cl, b_cl, prob);
  // 3) qkv = W_qkv @ x1
  {
    int O = 3 * C2_, C = C2_, tO = O / 32, tN = N_ / 64;
    gemm_wmma<<<B_ * tO * tN / 8, 256, 0, stream>>>(W_qkv, x1, qkvb, nullptr,
                                                    O, C, N_, tO, tN);
  }
  // 4) cluster-masked flash attention
  attn_kernel<<<B_ * HEADS_ * (N_ / 128), 256, 0, stream>>>(qkvb, prob, accb);
  // 5) out = W_proj @ acc + identity(x1)
  {
    int O = C2_, C = C2_, tO = O / 32, tN = N_ / 64;
    gemm_wmma<<<B_ * tO * tN / 8, 256, 0, stream>>>(W_proj, accb, out, x1,
                                                    O, C, N_, tO, tN);
  }
}